// TreeAttention_17008070492548
// MI455X (gfx1250) — compile-verified
//
#include <hip/hip_runtime.h>
#include <math.h>

// ---------------------------------------------------------------------------
// TreeAttention on MI455X (gfx1250, wave32). All matmuls use
// V_WMMA_F32_16X16X4_F32 (fp32 WMMA, matches fp32 reference numerics).
// Each wave computes a 32x64 C-tile: 2 M-frags x 4 N-frags = 8 independent
// accumulators -> 1.5 dwords/lane of L2 traffic per WMMA (vs 4 for 1 tile)
// and no serial WMMA->WMMA RAW chain.
//
// Fragment layouts per CDNA5 ISA 7.12.2 (wave32):
//   A (16x4):  row = lane&15; VGPR0 K = 2*(lane>>4), VGPR1 K = 2*(lane>>4)+1
//   B (4x16):  col = lane&15; VGPR0 K-row 2*(lane>>4), VGPR1 row 2*(lane>>4)+1
//   C (16x16): VGPR v -> row v + 8*(lane>>4), col lane&15
// ---------------------------------------------------------------------------

typedef float v2f __attribute__((ext_vector_type(2)));
typedef float v8f __attribute__((ext_vector_type(8)));

#define N_NODE 256
#define L_LEAF 2048
#define DF 256                 // d_feature == d_model == 256
#define NL (N_NODE + L_LEAF)   // 2304

#define WMMA_F32(acc, a, b) \
  __builtin_amdgcn_wmma_f32_16x16x4_f32(false, (a), false, (b), (short)0, (acc), false, false)

// ---- A-fragment loaders (fused element generation) --------------------------
struct ALoadPlain {
  const float* A; int ld;
  __device__ v2f frag(int row, int kk) const {
    const float* p = A + (size_t)row * ld + kk;
    v2f a; a.x = p[0]; a.y = p[1]; return a;
  }
};
struct ALoadMaskOmega {           // A[r,k] = (float)mask[r,k] * omega[k]
  const int* M; const float* omega; int ld;
  __device__ v2f frag(int row, int kk) const {
    const int* p = M + (size_t)row * ld + kk;
    v2f a;
    a.x = (float)p[0] * omega[kk];
    a.y = (float)p[1] * omega[kk + 1];
    return a;
  }
};
struct ALoadSoftmax {             // A[r,k] = exp(S[r,k]-cmax[k]) * cinv[k]
  const float* S; const float* cmax; const float* cinv; int ld;
  __device__ v2f frag(int row, int kk) const {
    const float* p = S + (size_t)row * ld + kk;
    v2f a;
    a.x = __expf(p[0] - cmax[kk])     * cinv[kk];
    a.y = __expf(p[1] - cmax[kk + 1]) * cinv[kk + 1];
    return a;
  }
};

// ---- GEMM: C[M,Nc] = genA[M,K] @ B[K,Nc] (+bias). 32x64 tile per wave. ------
template <class AL>
__global__ __launch_bounds__(32) void k_gemm_nn(AL al, const float* __restrict__ B,
    const float* __restrict__ bias, float* __restrict__ C, int K, int Nc) {
  const int m0 = blockIdx.x * 32, n0 = blockIdx.y * 64;
  const int lo = threadIdx.x & 15, h2 = (threadIdx.x >> 4) << 1;
  v8f acc[2][4] = {};
  for (int k = 0; k < K; k += 4) {
    const int kk = k + h2;
    const float* Bp = B + (size_t)kk * Nc + n0 + lo;
    __builtin_prefetch(Bp + 4 * Nc, 0, 1);           // next K-step (L2 prefetch)
    v2f a0 = al.frag(m0 + lo, kk);
    v2f a1 = al.frag(m0 + 16 + lo, kk);
    v2f b[4];
#pragma unroll
    for (int j = 0; j < 4; ++j) {
      b[j].x = Bp[16 * j];
      b[j].y = Bp[16 * j + Nc];
    }
#pragma unroll
    for (int j = 0; j < 4; ++j) {
      acc[0][j] = WMMA_F32(acc[0][j], a0, b[j]);
      acc[1][j] = WMMA_F32(acc[1][j], a1, b[j]);
    }
  }
#pragma unroll
  for (int i = 0; i < 2; ++i)
#pragma unroll
    for (int j = 0; j < 4; ++j) {
      const int col = n0 + 16 * j + lo;
      const float bb = bias ? bias[col] : 0.f;
#pragma unroll
      for (int v = 0; v < 8; ++v)
        C[(size_t)(m0 + 16 * i + v + (h2 << 2)) * Nc + col] = acc[i][j][v] + bb;
    }
}

// ---- GEMM: C[M,Nc] = scale * A[M,K] @ Bt[Nc,K]^T. 32x64 tile per wave. ------
__global__ __launch_bounds__(32) void k_gemm_abt(const float* __restrict__ A,
    const float* __restrict__ Bt, float* __restrict__ C, int K, int Nc, float scale) {
  const int m0 = blockIdx.x * 32, n0 = blockIdx.y * 64;
  const int lo = threadIdx.x & 15, h2 = (threadIdx.x >> 4) << 1;
  const float* Ar0 = A + (size_t)(m0 + lo) * K;
  const float* Ar1 = A + (size_t)(m0 + 16 + lo) * K;
  v8f acc[2][4] = {};
  for (int k = 0; k < K; k += 4) {
    const int kk = k + h2;
    __builtin_prefetch(Ar0 + kk + 16, 0, 1);
    v2f a0, a1, b[4];
    a0.x = Ar0[kk]; a0.y = Ar0[kk + 1];
    a1.x = Ar1[kk]; a1.y = Ar1[kk + 1];
#pragma unroll
    for (int j = 0; j < 4; ++j) {
      const float* Bp = Bt + (size_t)(n0 + 16 * j + lo) * K + kk;
      b[j].x = Bp[0]; b[j].y = Bp[1];
    }
#pragma unroll
    for (int j = 0; j < 4; ++j) {
      acc[0][j] = WMMA_F32(acc[0][j], a0, b[j]);
      acc[1][j] = WMMA_F32(acc[1][j], a1, b[j]);
    }
  }
#pragma unroll
  for (int i = 0; i < 2; ++i)
#pragma unroll
    for (int j = 0; j < 4; ++j) {
      const int col = n0 + 16 * j + lo;
#pragma unroll
      for (int v = 0; v < 8; ++v)
        C[(size_t)(m0 + 16 * i + v + (h2 << 2)) * Nc + col] = acc[i][j][v] * scale;
    }
}

// ---- omega[l] = leaf[l,:] . u_s ---------------------------------------------
__global__ void k_matvec(const float* __restrict__ leaf, const float* __restrict__ u,
                         float* __restrict__ omega, int D) {
  int l = blockIdx.x * blockDim.x + threadIdx.x;
  const float* row = leaf + (size_t)l * D;
  float s = 0.f;
  for (int d = 0; d < D; ++d) s += row[d] * u[d];
  omega[l] = s;
}

// ---- s1[n] = sum_l mask*omega ; cnt[n] = nnz(upward[n]) (exact) -------------
// Mask row (8KB) staged into LDS via the gfx1250 async-to-LDS path (ASYNCcnt).
__global__ __launch_bounds__(256) void k_stats(
    const int* __restrict__ mask, const float* __restrict__ omega,
    const float* __restrict__ nv, const float* __restrict__ lv,
    float* __restrict__ s1, float* __restrict__ cnt) {
  const int n = blockIdx.x, t = threadIdx.x;       // blockDim.x == 256 == DF
  __shared__ int msk[L_LEAF];
  __shared__ float red[256];
  const int* mrow = mask + (size_t)n * L_LEAF;
#pragma unroll
  for (int l = t; l < L_LEAF; l += 256) {
    unsigned long long g = (unsigned long long)(uintptr_t)(mrow + l);
    unsigned int       d = (unsigned int)(uintptr_t)(&msk[l]);
    asm volatile("global_load_async_to_lds_b32 %0, %1, off" :: "v"(d), "v"(g) : "memory");
  }
  asm volatile("s_wait_asynccnt 0x0" ::: "memory");
  __syncthreads();

  float s = 0.f;
  for (int l = t; l < L_LEAF; l += 256) s += msk[l] ? omega[l] : 0.f;
  red[t] = s; __syncthreads();
  for (int o = 128; o > 0; o >>= 1) { if (t < o) red[t] += red[t + o]; __syncthreads(); }
  if (t == 0) s1[n] = red[0];
  __syncthreads();

  const float nvd = nv[(size_t)n * DF + t];        // thread t owns feature d=t
  int c = 0;
  for (int l = 0; l < L_LEAF; ++l)
    if (msk[l]) c += ((nvd + lv[(size_t)l * DF + t]) != 0.f) ? 1 : 0;
  red[t] = (float)c; __syncthreads();
  for (int o = 128; o > 0; o >>= 1) { if (t < o) red[t] += red[t + o]; __syncthreads(); }
  if (t == 0) cnt[n] = red[0];
}

// ---- node_hat -> vals rows [0,N) --------------------------------------------
__global__ void k_finalize(const float* __restrict__ nv, const float* __restrict__ tb,
                           const float* __restrict__ s1, const float* __restrict__ cnt,
                           float* __restrict__ vals) {
  const int n = blockIdx.x, d = threadIdx.x;
  const float wa = 0.5f * (nv[(size_t)n * DF + d] * s1[n] + tb[(size_t)n * DF + d]);
  vals[(size_t)n * DF + d] = wa * ((float)DF / (cnt[n] + 1e-6f));
}

// ---- per-column (axis=-2) softmax stats, coalesced 64-col x 4-row tiles -----
__global__ __launch_bounds__(256) void k_colstats(const float* __restrict__ S, int R, int C,
                                                  float* __restrict__ cmax,
                                                  float* __restrict__ cinv) {
  const int cl = threadIdx.x & 63;                 // column within 64-wide strip
  const int rl = threadIdx.x >> 6;                 // 0..3 row lanes
  const int col = blockIdx.x * 64 + cl;
  __shared__ float red[4][64];
  __shared__ float smax[64];
  float m = -INFINITY;
  for (int r = rl; r < R; r += 4) m = fmaxf(m, S[(size_t)r * C + col]);
  red[rl][cl] = m; __syncthreads();
  if (rl == 0)
    smax[cl] = fmaxf(fmaxf(red[0][cl], red[1][cl]), fmaxf(red[2][cl], red[3][cl]));
  __syncthreads();
  const float mx = smax[cl];
  float s = 0.f;
  for (int r = rl; r < R; r += 4) s += __expf(S[(size_t)r * C + col] - mx);
  red[rl][cl] = s; __syncthreads();
  if (rl == 0) {
    cmax[col] = mx;
    cinv[col] = 1.f / (red[0][cl] + red[1][cl] + red[2][cl] + red[3][cl]);
  }
}

// ---------------------------------------------------------------------------
extern "C" void kernel_launch(void* const* d_in, const int* in_sizes, int n_in,
                              void* d_out, int out_size, void* d_ws, size_t ws_size,
                              hipStream_t stream) {
  (void)in_sizes; (void)n_in; (void)out_size; (void)ws_size;
  const float* node = (const float*)d_in[0];
  const float* leaf = (const float*)d_in[1];
  const int*   mask = (const int*)  d_in[2];
  const float* Wq   = (const float*)d_in[3];
  const float* bq   = (const float*)d_in[4];
  const float* Wk   = (const float*)d_in[5];
  const float* bk   = (const float*)d_in[6];
  const float* Wv   = (const float*)d_in[7];
  const float* bv   = (const float*)d_in[8];
  const float* u_s  = (const float*)d_in[9];

  float* ws = (float*)d_ws;
  size_t off = 0;
  float* nq    = ws + off; off += (size_t)N_NODE * DF;
  float* lq    = ws + off; off += (size_t)L_LEAF * DF;
  float* Kmat  = ws + off; off += (size_t)NL * DF;      // [nk; lk]
  float* nv    = ws + off; off += (size_t)N_NODE * DF;
  float* vals  = ws + off; off += (size_t)NL * DF;      // [node_hat; lv]
  float* omega = ws + off; off += L_LEAF;
  float* s1    = ws + off; off += N_NODE;
  float* cnt   = ws + off; off += N_NODE;
  float* tbuf  = ws + off; off += (size_t)N_NODE * DF;
  float* S1    = ws + off; off += (size_t)N_NODE * NL;
  float* S2    = ws + off; off += (size_t)L_LEAF * NL;
  float* cmax1 = ws + off; off += NL;
  float* cinv1 = ws + off; off += NL;
  float* cmax2 = ws + off; off += NL;
  float* cinv2 = ws + off; off += NL;

  float* lk = Kmat + (size_t)N_NODE * DF;
  float* lv = vals + (size_t)N_NODE * DF;

  const dim3 b32(32);
  // shared Q/K/V projections (WMMA f32), 32x64 tiles
  k_gemm_nn<ALoadPlain><<<dim3(8, 4),  b32, 0, stream>>>({node, DF}, Wq, bq, nq,   DF, DF);
  k_gemm_nn<ALoadPlain><<<dim3(8, 4),  b32, 0, stream>>>({node, DF}, Wk, bk, Kmat, DF, DF);
  k_gemm_nn<ALoadPlain><<<dim3(8, 4),  b32, 0, stream>>>({node, DF}, Wv, bv, nv,   DF, DF);
  k_gemm_nn<ALoadPlain><<<dim3(64, 4), b32, 0, stream>>>({leaf, DF}, Wq, bq, lq,   DF, DF);
  k_gemm_nn<ALoadPlain><<<dim3(64, 4), b32, 0, stream>>>({leaf, DF}, Wk, bk, lk,   DF, DF);
  k_gemm_nn<ALoadPlain><<<dim3(64, 4), b32, 0, stream>>>({leaf, DF}, Wv, bv, lv,   DF, DF);

  // omega = leaf @ u_s
  k_matvec<<<dim3(L_LEAF / 256), dim3(256), 0, stream>>>(leaf, u_s, omega, DF);

  // t = (mask * omega) @ lv   (A synthesized on the fly; WMMA)
  k_gemm_nn<ALoadMaskOmega><<<dim3(8, 4), b32, 0, stream>>>(
      {mask, omega, L_LEAF}, lv, nullptr, tbuf, L_LEAF, DF);

  // s1 = mask@omega, cnt = exact nnz of upward (mask staged via async-to-LDS)
  k_stats<<<dim3(N_NODE), dim3(256), 0, stream>>>(mask, omega, nv, lv, s1, cnt);

  // node_hat = 0.5*(nv*s1 + t) * (D/(cnt+1e-6)) -> vals rows [0,N)
  k_finalize<<<dim3(N_NODE), dim3(256), 0, stream>>>(nv, tbuf, s1, cnt, vals);

  // scores: S1 = nq @ K^T * scale  [N, N+L];  S2 = lq @ K^T * scale [L, N+L]
  const float scale = 0.0625f;  // 1/sqrt(256)
  k_gemm_abt<<<dim3(8, NL / 64),  b32, 0, stream>>>(nq, Kmat, S1, DF, NL, scale);
  k_gemm_abt<<<dim3(64, NL / 64), b32, 0, stream>>>(lq, Kmat, S2, DF, NL, scale);

  // column-wise (axis=-2) softmax stats, coalesced
  k_colstats<<<dim3(NL / 64), dim3(256), 0, stream>>>(S1, N_NODE, NL, cmax1, cinv1);
  k_colstats<<<dim3(NL / 64), dim3(256), 0, stream>>>(S2, L_LEAF, NL, cmax2, cinv2);

  // outputs: Attn = softmax_col(S1) @ vals ; Attl = softmax_col(S2) @ vals
  float* Attn = (float*)d_out;
  float* Attl = Attn + (size_t)N_NODE * DF;
  k_gemm_nn<ALoadSoftmax><<<dim3(8, 4),  b32, 0, stream>>>(
      {S1, cmax1, cinv1, NL}, vals, nullptr, Attn, NL, DF);
  k_gemm_nn<ALoadSoftmax><<<dim3(64, 4), b32, 0, stream>>>(
      {S2, cmax2, cinv2, NL}, vals, nullptr, Attl, NL, DF);
}